// BaseGNN_18545668784843
// MI455X (gfx1250) — compile-verified
//
#include <hip/hip_runtime.h>

#define N_NODES 50000
#define N_EDGES 1000000
#define F_IN    16
#define HID     128
#define HID2    64
#define N_CLS   72
#define BN_EPS  1e-5f

typedef float v2f __attribute__((ext_vector_type(2)));
typedef float v8f __attribute__((ext_vector_type(8)));

// ---------------- degree / normalization ----------------

__global__ void deg_init_k(float* deg, int n) {
    int i = blockIdx.x * blockDim.x + threadIdx.x;
    if (i < n) deg[i] = 1.0f;               // self-loop contributes 1
}

__global__ void deg_acc_k(const int* __restrict__ dst, float* deg, int e) {
    int i = blockIdx.x * blockDim.x + threadIdx.x;
    if (i < e) atomicAdd(&deg[dst[i]], 1.0f);
}

__global__ void deg_inv_k(float* deg, int n) {
    int i = blockIdx.x * blockDim.x + threadIdx.x;
    if (i < n) {
        float d = deg[i];
        deg[i] = (d > 0.0f) ? rsqrtf(d) : 0.0f;
    }
}

// ---------------- fp32 WMMA GEMM: C = A[MxK] @ B[KxN] (+bias, relu) ----------------
// One wave computes a 16-row strip across all NT column tiles.
// B is staged in LDS (zero-padded to NT*16 columns).

template <int NT>
__global__ void gemm_wmma_k(const float* __restrict__ A,
                            const float* __restrict__ B,
                            const float* __restrict__ bias,
                            float* __restrict__ C,
                            int M, int K, int Nreal, int relu) {
    extern __shared__ float bs[];           // K x (NT*16)
    const int ldb = NT * 16;

    // cooperative load of B into LDS (pad columns with zero)
    for (int idx = threadIdx.x; idx < K * ldb; idx += blockDim.x) {
        int col = idx % ldb;
        int row = idx / ldb;
        bs[idx] = (col < Nreal) ? B[row * Nreal + col] : 0.0f;
    }
    __syncthreads();

    const int lane   = threadIdx.x & 31;
    const int wave   = (blockIdx.x * blockDim.x + threadIdx.x) >> 5;
    const int tile_m = wave;
    if (tile_m * 16 >= M) return;           // whole wave uniform -> EXEC stays full

    const int half = lane >> 4;             // 0: K0/K1, 1: K2/K3
    const int lidx = lane & 15;

    const float* arow = A + (size_t)(tile_m * 16 + lidx) * K;

    v8f acc[NT] = {};

    for (int k = 0; k < K; k += 4) {
        v2f a;
        a.x = arow[k + 2 * half + 0];
        a.y = arow[k + 2 * half + 1];
#pragma unroll
        for (int t = 0; t < NT; ++t) {
            const int ncol = t * 16 + lidx;
            v2f b;
            b.x = bs[(k + 2 * half + 0) * ldb + ncol];
            b.y = bs[(k + 2 * half + 1) * ldb + ncol];
            acc[t] = __builtin_amdgcn_wmma_f32_16x16x4_f32(
                false, a, false, b, (short)0, acc[t], false, false);
        }
    }

#pragma unroll
    for (int t = 0; t < NT; ++t) {
        const int col = t * 16 + lidx;
        if (col >= Nreal) continue;
        const float bv = bias ? bias[col] : 0.0f;
#pragma unroll
        for (int r = 0; r < 8; ++r) {
            const int row = tile_m * 16 + r + 8 * half;
            float v = acc[t][r] + bv;
            if (relu) v = fmaxf(v, 0.0f);
            C[(size_t)row * Nreal + col] = v;
        }
    }
}

// ---------------- GCN aggregation ----------------

// agg[n,f] = hw[n,f] * dinv[n]^2 + bias[f]   (self-loop term + conv bias)
__global__ void self_init_k(const float* __restrict__ hw,
                            const float* __restrict__ dinv,
                            const float* __restrict__ bias,
                            float* __restrict__ agg, int total) {
    int i = blockIdx.x * blockDim.x + threadIdx.x;
    if (i >= total) return;
    int n = i >> 7;           // / HID
    int f = i & (HID - 1);
    float di = dinv[n];
    agg[i] = hw[i] * di * di + bias[f];
}

// one wave per edge; lane handles one float4 of the 128-wide feature row
__global__ void edge_scatter_k(const int* __restrict__ srcs,
                               const int* __restrict__ dsts,
                               const float* __restrict__ dinv,
                               const float* __restrict__ hw,
                               float* __restrict__ agg, int E) {
    int t = blockIdx.x * blockDim.x + threadIdx.x;
    int e = t >> 5;
    int lane = t & 31;
    if (e >= E) return;
    int s = srcs[e];
    int d = dsts[e];
    float w = dinv[s] * dinv[d];
    const float4* row = (const float4*)(hw + (size_t)s * HID);
    float4 v = row[lane];
    float* out = agg + (size_t)d * HID + lane * 4;
    atomicAdd(out + 0, v.x * w);
    atomicAdd(out + 1, v.y * w);
    atomicAdd(out + 2, v.z * w);
    atomicAdd(out + 3, v.w * w);
}

// ---------------- batchnorm ----------------

__global__ void col_sum_k(const float* __restrict__ X, float* __restrict__ sum, int rows) {
    int col = threadIdx.x & (HID - 1);
    int sub = threadIdx.x >> 7;                 // 0..1 for blockDim 256
    int rpi = blockDim.x >> 7;                  // rows per iteration
    float acc = 0.0f;
    for (int r = blockIdx.x * rpi + sub; r < rows; r += gridDim.x * rpi)
        acc += X[(size_t)r * HID + col];
    atomicAdd(&sum[col], acc);
}

__global__ void col_var_k(const float* __restrict__ X, const float* __restrict__ mean,
                          float* __restrict__ sum, int rows) {
    int col = threadIdx.x & (HID - 1);
    int sub = threadIdx.x >> 7;
    int rpi = blockDim.x >> 7;
    float mu = mean[col];
    float acc = 0.0f;
    for (int r = blockIdx.x * rpi + sub; r < rows; r += gridDim.x * rpi) {
        float d = X[(size_t)r * HID + col] - mu;
        acc += d * d;
    }
    atomicAdd(&sum[col], acc);
}

__global__ void mean_fin_k(const float* sum, float* mean, float inv_n) {
    int i = threadIdx.x;
    mean[i] = sum[i] * inv_n;
}

__global__ void rstd_fin_k(const float* sum, float* rstd, float inv_n) {
    int i = threadIdx.x;
    rstd[i] = rsqrtf(sum[i] * inv_n + BN_EPS);
}

// h = (residual ? h : 0) + relu((agg - mean) * rstd * g + b)
__global__ void bn_apply_k(const float* __restrict__ agg,
                           const float* __restrict__ mean,
                           const float* __restrict__ rstd,
                           const float* __restrict__ g,
                           const float* __restrict__ bb,
                           float* __restrict__ h, int total, int residual) {
    int i = blockIdx.x * blockDim.x + threadIdx.x;
    if (i >= total) return;
    int f = i & (HID - 1);
    float v = (agg[i] - mean[f]) * rstd[f] * g[f] + bb[f];
    v = fmaxf(v, 0.0f);
    h[i] = residual ? (h[i] + v) : v;
}

// ---------------- orchestration ----------------

extern "C" void kernel_launch(void* const* d_in, const int* in_sizes, int n_in,
                              void* d_out, int out_size, void* d_ws, size_t ws_size,
                              hipStream_t stream) {
    const float* x      = (const float*)d_in[0];
    const int*   ei     = (const int*)d_in[1];      // int32: JAX x64-disabled
    const int*   e_src  = ei;                       // edge_index[0]
    const int*   e_dst  = ei + N_EDGES;             // edge_index[1]
    const float* W_enc  = (const float*)d_in[2];
    const float* b_enc  = (const float*)d_in[3];
    const float* W_cls1 = (const float*)d_in[4];
    const float* b_cls1 = (const float*)d_in[5];
    const float* W_cls2 = (const float*)d_in[6];
    const float* b_cls2 = (const float*)d_in[7];

    // workspace carve-out (256B aligned)
    char*  ws  = (char*)d_ws;
    size_t off = 0;
    auto carve = [&](size_t nbytes) -> float* {
        float* p = (float*)(ws + off);
        off += (nbytes + 255) & ~(size_t)255;
        return p;
    };
    float* dinv = carve((size_t)N_NODES * 4);
    float* h    = carve((size_t)N_NODES * HID * 4);
    float* hw   = carve((size_t)N_NODES * HID * 4);
    float* agg  = carve((size_t)N_NODES * HID * 4);
    float* hc   = carve((size_t)N_NODES * HID2 * 4);
    float* ssum = carve((size_t)HID * 4);
    float* mean = carve((size_t)HID * 4);
    float* rstd = carve((size_t)HID * 4);
    (void)ws_size; (void)in_sizes; (void)n_in; (void)out_size;

    // --- symmetric GCN normalization ---
    deg_init_k<<<(N_NODES + 255) / 256, 256, 0, stream>>>(dinv, N_NODES);
    deg_acc_k<<<(N_EDGES + 255) / 256, 256, 0, stream>>>(e_dst, dinv, N_EDGES);
    deg_inv_k<<<(N_NODES + 255) / 256, 256, 0, stream>>>(dinv, N_NODES);

    const int tiles_m  = N_NODES / 16;         // 3125 exact
    const int gblocks  = (tiles_m + 3) / 4;    // 4 waves / block (blockDim 128)
    const int ew_total = (N_NODES * HID + 255) / 256;

    // --- encoder: h = relu(x @ W_enc + b_enc) ---
    gemm_wmma_k<8><<<gblocks, 128, F_IN * HID * 4, stream>>>(
        x, W_enc, b_enc, h, N_NODES, F_IN, HID, 1);

    // --- 3 GCN conv layers ---
    for (int l = 0; l < 3; ++l) {
        const float* Wc = (const float*)d_in[8 + 4 * l];
        const float* bc = (const float*)d_in[9 + 4 * l];
        const float* g  = (const float*)d_in[10 + 4 * l];
        const float* bb = (const float*)d_in[11 + 4 * l];

        gemm_wmma_k<8><<<gblocks, 128, HID * HID * 4, stream>>>(
            h, Wc, nullptr, hw, N_NODES, HID, HID, 0);

        self_init_k<<<ew_total, 256, 0, stream>>>(hw, dinv, bc, agg, N_NODES * HID);
        edge_scatter_k<<<(N_EDGES * 32) / 256, 256, 0, stream>>>(
            e_src, e_dst, dinv, hw, agg, N_EDGES);

        hipMemsetAsync(ssum, 0, HID * 4, stream);
        col_sum_k<<<256, 256, 0, stream>>>(agg, ssum, N_NODES);
        mean_fin_k<<<1, HID, 0, stream>>>(ssum, mean, 1.0f / N_NODES);

        hipMemsetAsync(ssum, 0, HID * 4, stream);
        col_var_k<<<256, 256, 0, stream>>>(agg, mean, ssum, N_NODES);
        rstd_fin_k<<<1, HID, 0, stream>>>(ssum, rstd, 1.0f / N_NODES);

        bn_apply_k<<<ew_total, 256, 0, stream>>>(
            agg, mean, rstd, g, bb, h, N_NODES * HID, l > 0);
    }

    // --- classifier ---
    gemm_wmma_k<4><<<gblocks, 128, HID * (4 * 16) * 4, stream>>>(
        h, W_cls1, b_cls1, hc, N_NODES, HID, HID2, 1);
    gemm_wmma_k<5><<<gblocks, 128, HID2 * (5 * 16) * 4, stream>>>(
        hc, W_cls2, b_cls2, (float*)d_out, N_NODES, HID2, N_CLS, 0);
}